// NevenLoss_80882824118610
// MI455X (gfx1250) — compile-verified
//
#include <hip/hip_runtime.h>

// Problem constants (match reference)
#define B_ 8
#define C_ 2
#define H_ 192
#define W_ 224
#define P_ (H_ * W_)          // 43008 pixels per (b,c) map
#define I_ 64                 // instances
#define TILES_PER_MAP (P_ / 16)   // 2688
#define T_TOTAL (B_ * C_ * TILES_PER_MAP)  // 43008 tiles

typedef __attribute__((ext_vector_type(2))) float v2f;
typedef __attribute__((ext_vector_type(8))) float v8f;

// -------------------------------------------------------------------------
// Phase 1: per (b,c) segmented sums via LDS float atomics (ds_add_f32).
// Produces, per (b,c,i), the B-matrix column for the WMMA:
//   [ 2*inv*cy, 2*inv*cx, -inv, -inv*c2 ]   (inv = 1/(2*sig_mean^2))
// and accumulates the smooth loss directly into the scalar output.
// -------------------------------------------------------------------------
__global__ __launch_bounds__(256) void neven_phase1(
    const float* __restrict__ off,   // [B,2,P]
    const float* __restrict__ sig,   // [B,1,P]
    const int*   __restrict__ lab,   // [B,C,P]
    float4*      __restrict__ bcols, // [B*C, I]
    float*       __restrict__ out)   // scalar accumulator
{
    __shared__ float cnt[I_], sy[I_], sx[I_], ss[I_], ss2[I_];
    const int bc  = blockIdx.x;          // 0..15
    const int b   = bc / C_;
    const int tid = threadIdx.x;

    for (int i = tid; i < I_; i += blockDim.x) {
        cnt[i] = 0.f; sy[i] = 0.f; sx[i] = 0.f; ss[i] = 0.f; ss2[i] = 0.f;
    }
    __syncthreads();

    const int*   labp = lab + bc * P_;
    const float* offy = off + (b * 2 + 0) * P_;
    const float* offx = off + (b * 2 + 1) * P_;
    const float* sgp  = sig + b * P_;

    for (int p = tid; p < P_; p += blockDim.x) {
        const int l = labp[p];
        if (l > 0) {
            const int gy = p / W_;
            const int gx = p - gy * W_;
            const float ey = (gy * (64.0f / (float)H_) + offy[p]) * (1.0f / 64.0f);
            const float ex = (gx * (64.0f / (float)W_) + offx[p]) * (1.0f / 64.0f);
            // sigma = SCALE / (1 + exp(-x / (2*SCALE)))
            const float sg = 64.0f / (1.0f + __expf(-sgp[p] * (1.0f / 128.0f)));
            const int i = l - 1;
            atomicAdd(&cnt[i], 1.0f);     // ds_add_f32
            atomicAdd(&sy[i], ey);
            atomicAdd(&sx[i], ex);
            atomicAdd(&ss[i], sg);
            atomicAdd(&ss2[i], sg * sg);
        }
    }
    __syncthreads();

    if (tid < I_) {
        const float n   = cnt[tid];
        const float cy  = sy[tid] / n;
        const float cx  = sx[tid] / n;
        const float m   = ss[tid] / n;           // sig_mean
        const float inv = 1.0f / (2.0f * m * m);
        const float c2  = cy * cy + cx * cx;
        bcols[bc * I_ + tid] =
            make_float4(2.0f * inv * cy, 2.0f * inv * cx, -inv, -inv * c2);
        // smooth_per = (ss2 - 2*m*ss + n*m^2)/n ; scaled by SW/(N*B*C) = 1/16384
        const float smooth = (ss2[tid] - 2.0f * m * ss[tid] + n * m * m) / n;
        atomicAdd(out, smooth * (1.0f / 16384.0f));
    }
}

// -------------------------------------------------------------------------
// Phase 2: one wave per 16-pixel tile. V_WMMA_F32_16X16X4_F32 computes a
// 16(pixel) x 16(instance) tile of exp-arguments; 4 WMMAs cover all 64
// instances. VALU does exp / (phi-mask)^2 / seed regression.
// A layout (ISA 16x4 f32): lanes 0-15 -> K0,K1 = (ey,ex); lanes 16-31 ->
// K2,K3 = (e2, 1). B column per instance from bcols.
// D layout: lane L, vgpr r -> pixel row (L<16 ? r : 8+r), instance (L&15).
// -------------------------------------------------------------------------
__global__ __launch_bounds__(256) void neven_phase2(
    const float* __restrict__ seed,  // [B,C,P]
    const float* __restrict__ off,   // [B,2,P]
    const int*   __restrict__ lab,   // [B,C,P]
    const float4* __restrict__ bcols,// [B*C, I]
    float*       __restrict__ out)   // scalar accumulator
{
    const int lane   = threadIdx.x & 31;
    const int waveId = (blockIdx.x * blockDim.x + threadIdx.x) >> 5;
    const int nWaves = (gridDim.x * blockDim.x) >> 5;
    const int half   = (lane >= 16) ? 1 : 0;
    const int lrow   = lane & 15;

    float inst_acc = 0.0f;   // sum (phi - mask)^2
    float seed_acc = 0.0f;   // sum (seed - seed_mask)^2, expanded

    for (int t = waveId; t < T_TOTAL; t += nWaves) {
        const int bc   = t / TILES_PER_MAP;
        const int tile = t - bc * TILES_PER_MAP;
        const int b    = bc / C_;
        const int p0   = tile * 16;

        const float* offy  = off  + (b * 2 + 0) * P_;
        const float* offx  = off  + (b * 2 + 1) * P_;
        const int*   labp  = lab  + bc * P_;
        const float* seedp = seed + bc * P_;
        const float4* bp   = bcols + bc * I_;

        // Each lane owns pixel row (lane&15) of the tile.
        const int   p  = p0 + lrow;
        const int   gy = p / W_;
        const int   gx = p - gy * W_;
        const float ey = (gy * (64.0f / (float)H_) + offy[p]) * (1.0f / 64.0f);
        const float ex = (gx * (64.0f / (float)W_) + offx[p]) * (1.0f / 64.0f);
        const int   mylab  = labp[p];
        const float myseed = seedp[p];

        // A operand: K0,K1 in lanes 0-15 ; K2,K3 (e2, 1) in lanes 16-31.
        v2f a;
        if (!half) { a.x = ey;              a.y = ex;   }
        else       { a.x = ey * ey + ex * ex; a.y = 1.0f; }

        // Gather the 8 labels/seeds this lane's D registers refer to.
        int   labv[8];
        float seedv[8];
        const int sbase = half ? 8 : 0;
#pragma unroll
        for (int r = 0; r < 8; ++r) {
            labv[r]  = __shfl(mylab,  sbase + r, 32);
            seedv[r] = __shfl(myseed, sbase + r, 32);
        }

        // seed^2 baseline, once per pixel (lane&15)==0 lanes cover all 16 rows.
        if (lrow == 0) {
#pragma unroll
            for (int r = 0; r < 8; ++r) seed_acc += seedv[r] * seedv[r];
        }

#pragma unroll
        for (int g = 0; g < 4; ++g) {
            // B operand: column = instance g*16 + (lane&15); lanes 0-15 take
            // rows K0,K1, lanes 16-31 take rows K2,K3.
            const float4 col = bp[g * 16 + lrow];
            v2f bb;
            if (!half) { bb.x = col.x; bb.y = col.y; }
            else       { bb.x = col.z; bb.y = col.w; }

            v8f c = {};
            // D[pixel, instance] = -dist2 * inv  (argument of exp)
            v8f d = __builtin_amdgcn_wmma_f32_16x16x4_f32(
                false, a, false, bb, (short)0, c, false, false);

            const int inst = g * 16 + lrow + 1;   // instance id (1-based)
#pragma unroll
            for (int r = 0; r < 8; ++r) {
                const float phi  = __expf(d[r]);
                const float msk  = (labv[r] == inst) ? 1.0f : 0.0f;
                const float diff = phi - msk;
                inst_acc += diff * diff;
                // seed cross terms only where mask==1 (one-hot over instances)
                seed_acc += msk * phi * (phi - 2.0f * seedv[r]);
            }
        }
    }

    // Wave reduction then one global atomic per wave.
#pragma unroll
    for (int o = 16; o > 0; o >>= 1) {
        inst_acc += __shfl_down(inst_acc, o, 32);
        seed_acc += __shfl_down(seed_acc, o, 32);
    }
    if (lane == 0) {
        // IW*inst/(N*BC) = inst/16384 ; SDW*seed/(P*BC) = seed/688128
        atomicAdd(out, inst_acc * (1.0f / 16384.0f) +
                       seed_acc * (1.0f / 688128.0f));
    }
}

// -------------------------------------------------------------------------
extern "C" void kernel_launch(void* const* d_in, const int* in_sizes, int n_in,
                              void* d_out, int out_size, void* d_ws, size_t ws_size,
                              hipStream_t stream) {
    const float* seed = (const float*)d_in[0];   // [B,C,H,W]
    const float* off  = (const float*)d_in[1];   // [B,2,H,W]
    const float* sig  = (const float*)d_in[2];   // [B,1,H,W]
    const int*   lab  = (const int*)  d_in[3];   // [B,C,H,W]
    float* out = (float*)d_out;                  // scalar f32
    float4* bcols = (float4*)d_ws;               // [B*C, I] = 16 KB

    hipMemsetAsync(out, 0, sizeof(float), stream);

    neven_phase1<<<B_ * C_, 256, 0, stream>>>(off, sig, lab, bcols, out);

    // 1344 blocks * 8 waves = 10752 waves -> exactly 4 tiles per wave
    neven_phase2<<<1344, 256, 0, stream>>>(seed, off, lab, bcols, out);

    (void)in_sizes; (void)n_in; (void)out_size; (void)ws_size;
}